// DetailBranch_72937134620862
// MI455X (gfx1250) — compile-verified
//
#include <hip/hip_runtime.h>

// ---------------------------------------------------------------------------
// Fused guided-filter for MI455X (gfx1250, wave32).
// One workgroup = one 32x32 output tile, 64x64 input region (16-px halo =
// 8 for stage-1 box means + 8 for stage-2 box means). All intermediates in
// LDS; 17-tap zero-padded box sums are done as banded GEMMs on the matrix
// pipe via V_WMMA_F32_16X16X4_F32 (f32 precision == reference).
// ---------------------------------------------------------------------------

typedef float v2f __attribute__((ext_vector_type(2)));
typedef float v8f __attribute__((ext_vector_type(8)));

#define BLOCK  128
#define NWAVES 4

// Band matrix value: Band[kk, n] = 1 iff the 17-wide window centered at
// (n + 8) covers relative column kk, i.e. n <= kk <= n + 16.
__device__ __forceinline__ float bandv(int kk, int n) {
  return (kk >= n && kk <= n + 16) ? 1.0f : 0.0f;
}

__device__ __forceinline__ v8f wmma4(v2f a, v2f b, v8f c) {
  // (neg_a, A, neg_b, B, c_mod, C, reuse_a, reuse_b)
  return __builtin_amdgcn_wmma_f32_16x16x4_f32(
      false, a, false, b, (short)0, c, false, false);
}

// Horizontal 17-tap box SUM: dst[r, c] = sum_{d=0..16} src[r, c+d]
// (output col c corresponds to window center c+8 in src coords).
// src must be at least (ncols_out + 16) wide.
__device__ void hpass(const float* src, int sstr, int nrows, int ncols_out,
                      float* dst, int dstr, int lane, int wid) {
  const int m  = lane & 15;
  const int hi = lane >> 4;
  const int tc = ncols_out >> 4, tr = nrows >> 4;
  for (int t = wid; t < tr * tc; t += NWAVES) {
    const int r0 = (t / tc) << 4;
    const int c0 = (t % tc) << 4;
    v8f acc = {};
    for (int ks = 0; ks < 8; ++ks) {          // K = 32 in steps of 4
      const int kk = ks * 4 + hi * 2;         // relative K index this lane-half
      v2f a, b;
      a.x = src[(r0 + m) * sstr + c0 + kk];       // A: 16x4 image tile
      a.y = src[(r0 + m) * sstr + c0 + kk + 1];
      b.x = bandv(kk, m);                         // B: 4x16 band tile
      b.y = bandv(kk + 1, m);
      acc = wmma4(a, b, acc);
    }
    for (int v = 0; v < 8; ++v)
      dst[(r0 + v + hi * 8) * dstr + c0 + m] = acc[v];
  }
}

// Vertical 17-tap box SUM (x scale): dst[r, c] = scale * sum_{d=0..16} src[r+d, c]
// src must have at least (nrows_out + 16) rows.
__device__ void vpass(const float* src, int sstr, int nrows_out, int ncols,
                      float* dst, int dstr, float scale, int lane, int wid) {
  const int m  = lane & 15;
  const int hi = lane >> 4;
  const int tc = ncols >> 4, tr = nrows_out >> 4;
  for (int t = wid; t < tr * tc; t += NWAVES) {
    const int r0 = (t / tc) << 4;
    const int c0 = (t % tc) << 4;
    v8f acc = {};
    for (int ks = 0; ks < 8; ++ks) {
      const int kk = ks * 4 + hi * 2;
      v2f a, b;
      a.x = bandv(kk, m);                          // A: 16x4 band^T tile
      a.y = bandv(kk + 1, m);
      b.x = src[(r0 + kk) * sstr + c0 + m];        // B: 4x16 image tile
      b.y = src[(r0 + kk + 1) * sstr + c0 + m];
      acc = wmma4(a, b, acc);
    }
    for (int v = 0; v < 8; ++v)
      dst[(r0 + v + hi * 8) * dstr + c0 + m] = acc[v] * scale;
  }
}

__global__ __launch_bounds__(BLOCK)
void guided_detail_kernel(const float* __restrict__ x, float* __restrict__ out) {
  extern __shared__ float smem[];
  // LDS plan (floats):
  float* sG  = smem;             // 3 * 4096 : x_norm channels, 64x64 region
  float* sI  = sG  + 3 * 4096;   // 4096     : gray, 64x64
  float* sQ  = sI  + 4096;       // 4096     : scratch quantity plane
  float* sH  = sQ  + 4096;       // 3072     : horizontal sums (64x48 / 48x32)
  float* mI  = sH  + 3072;       // 2304     : mean_i, 48x48
  float* mG  = mI  + 2304;       // 2304
  float* mGI = mG  + 2304;       // 2304
  float* mGG = mGI + 2304;       // 2304
  float* aP  = mGG + 2304;       // 2304
  float* bP  = aP  + 2304;       // 2304
  float* mA  = bP  + 2304;       // 1024     : mean_a, 32x32
  float* mB  = mA  + 1024;       // 1024     : mean_b, 32x32

  const int tid  = threadIdx.x;
  const int lane = tid & 31;
  const int wid  = tid >> 5;
  const int bx = blockIdx.x * 32, by = blockIdx.y * 32;
  const int batch = blockIdx.z;
  const int x0 = bx - 16, y0 = by - 16;   // region origin (with halo)

  const float cmean[3] = {0.485f, 0.456f, 0.406f};
  const float cstd[3]  = {0.229f, 0.224f, 0.225f};

  // ---- Phase 0: load + denormalize + clip; zero outside image (== zero pad)
  for (int idx = tid; idx < 64 * 64; idx += BLOCK) {
    const int ly = idx >> 6, lx = idx & 63;
    const int gy = y0 + ly, gx = x0 + lx;
    const bool inb = (gy >= 0) & (gy < 512) & (gx >= 0) & (gx < 512);
    float s = 0.0f;
    for (int c = 0; c < 3; ++c) {
      float v = 0.0f;
      if (inb) {
        v = x[(((size_t)batch * 3 + c) * 512 + gy) * 512 + gx];
        v = v * cstd[c] + cmean[c];
        v = fminf(fmaxf(v, 0.0f), 1.0f);
      }
      sG[c * 4096 + idx] = v;
      s += v;
    }
    sI[idx] = s * (1.0f / 3.0f);
  }
  __syncthreads();

  const float inv289 = 1.0f / 289.0f;

  // ---- mean_i (shared across channels): 64x64 -> 48x48
  hpass(sI, 64, 64, 48, sH, 48, lane, wid);
  __syncthreads();
  vpass(sH, 48, 48, 48, mI, 48, inv289, lane, wid);
  __syncthreads();

  for (int c = 0; c < 3; ++c) {
    const float* g = sG + c * 4096;

    // mean_g
    hpass(g, 64, 64, 48, sH, 48, lane, wid);
    __syncthreads();
    vpass(sH, 48, 48, 48, mG, 48, inv289, lane, wid);
    __syncthreads();

    // mean_gi
    for (int idx = tid; idx < 4096; idx += BLOCK) sQ[idx] = g[idx] * sI[idx];
    __syncthreads();
    hpass(sQ, 64, 64, 48, sH, 48, lane, wid);
    __syncthreads();
    vpass(sH, 48, 48, 48, mGI, 48, inv289, lane, wid);
    __syncthreads();

    // mean_gg
    for (int idx = tid; idx < 4096; idx += BLOCK) sQ[idx] = g[idx] * g[idx];
    __syncthreads();
    hpass(sQ, 64, 64, 48, sH, 48, lane, wid);
    __syncthreads();
    vpass(sH, 48, 48, 48, mGG, 48, inv289, lane, wid);
    __syncthreads();

    // a, b on the 48x48 halo region (zero outside image falls out: 0/eps = 0)
    for (int idx = tid; idx < 48 * 48; idx += BLOCK) {
      const float mg = mG[idx], mi = mI[idx];
      const float cov = mGI[idx] - mg * mi;
      const float var = mGG[idx] - mg * mg;
      const float a = cov / (var + 0.001f);
      aP[idx] = a;
      bP[idx] = mi - a * mg;
    }
    __syncthreads();

    // mean_a: 48x48 -> 32x32
    hpass(aP, 48, 48, 32, sH, 32, lane, wid);
    __syncthreads();
    vpass(sH, 32, 32, 32, mA, 32, inv289, lane, wid);
    __syncthreads();

    // mean_b: 48x48 -> 32x32
    hpass(bP, 48, 48, 32, sH, 32, lane, wid);
    __syncthreads();
    vpass(sH, 32, 32, 32, mB, 32, inv289, lane, wid);
    __syncthreads();

    // out = x_norm - (mean_a * x_norm + mean_b)
    for (int idx = tid; idx < 32 * 32; idx += BLOCK) {
      const int r = idx >> 5, cc = idx & 31;
      const float gg = g[(16 + r) * 64 + (16 + cc)];
      const float res = gg - (mA[idx] * gg + mB[idx]);
      out[(((size_t)batch * 3 + c) * 512 + (by + r)) * 512 + (bx + cc)] = res;
    }
    __syncthreads();  // mA/mB/sH/means reused next channel
  }
}

extern "C" void kernel_launch(void* const* d_in, const int* in_sizes, int n_in,
                              void* d_out, int out_size, void* d_ws, size_t ws_size,
                              hipStream_t stream) {
  (void)in_sizes; (void)n_in; (void)d_ws; (void)ws_size; (void)out_size;
  const float* x = (const float*)d_in[0];
  float* out = (float*)d_out;

  const size_t smem_bytes =
      (3 * 4096 + 4096 + 4096 + 3072 + 6 * 2304 + 2 * 1024) * sizeof(float); // ~154 KB

  // Opt in to >64KB dynamic LDS (gfx1250 WGP supports 320KB). Deterministic,
  // host-side only; safe under graph capture. Ignore error if unsupported.
  (void)hipFuncSetAttribute((const void*)guided_detail_kernel,
                            hipFuncAttributeMaxDynamicSharedMemorySize,
                            (int)smem_bytes);

  dim3 grid(512 / 32, 512 / 32, 16);  // x tiles, y tiles, batch
  dim3 block(BLOCK);
  guided_detail_kernel<<<grid, block, smem_bytes, stream>>>(x, out);
}